// GCNModel_42863773614468
// MI455X (gfx1250) — compile-verified
//
#include <hip/hip_runtime.h>
#include <hip/hip_bf16.h>

typedef float v2f __attribute__((ext_vector_type(2)));
typedef float v8f __attribute__((ext_vector_type(8)));

// ---------------------------------------------------------------------------
// Zero-fill (graph-replay safe: all accumulators re-zeroed every launch)
// ---------------------------------------------------------------------------
__global__ void fill_zero_kernel(float* __restrict__ p, long long n) {
    long long i = (long long)blockIdx.x * blockDim.x + threadIdx.x;
    if (i < n) p[i] = 0.0f;
}

// ---------------------------------------------------------------------------
// Degree counting: out_deg[src[e]]++, in_deg[dst[e]]++
// ---------------------------------------------------------------------------
__global__ void degree_kernel(const int* __restrict__ src, const int* __restrict__ dst,
                              float* __restrict__ odeg, float* __restrict__ ideg, int E) {
    int e = blockIdx.x * blockDim.x + threadIdx.x;
    if (e >= E) return;
    atomicAdd(&odeg[src[e]], 1.0f);
    atomicAdd(&ideg[dst[e]], 1.0f);
}

// deg -> rsqrt(max(deg,1)), in place
__global__ void norm_kernel(float* __restrict__ onorm, float* __restrict__ inorm, int N) {
    int i = blockIdx.x * blockDim.x + threadIdx.x;
    if (i >= N) return;
    onorm[i] = rsqrtf(fmaxf(onorm[i], 1.0f));
    inorm[i] = rsqrtf(fmaxf(inorm[i], 1.0f));
}

// x_scaled = h * out_norm (row broadcast); feature dim fixed at 128
__global__ void scale_kernel(const float* __restrict__ h, const float* __restrict__ onorm,
                             float* __restrict__ out, long long NF) {
    long long i = (long long)blockIdx.x * blockDim.x + threadIdx.x;
    if (i >= NF) return;
    out[i] = h[i] * onorm[i >> 7];
}

// ---------------------------------------------------------------------------
// Edge aggregation: one wave32 per edge, 4 floats per lane (float4 gather,
// 4x global_atomic_add_f32 scatter). Feature matrices fit in 192MB L2, so
// this is L2-resident traffic.
// ---------------------------------------------------------------------------
__global__ __launch_bounds__(256) void aggregate_kernel(
    const float* __restrict__ x, const int* __restrict__ src, const int* __restrict__ dst,
    float* __restrict__ agg, int E) {
    int wave = (int)(((long long)blockIdx.x * blockDim.x + threadIdx.x) >> 5);
    if (wave >= E) return;
    int lane = threadIdx.x & 31;
    int s = src[wave];
    int d = dst[wave];
    const float4 v = *(const float4*)(x + (size_t)s * 128 + lane * 4);
    float* b = agg + (size_t)d * 128 + lane * 4;
    atomicAdd(b + 0, v.x);
    atomicAdd(b + 1, v.y);
    atomicAdd(b + 2, v.z);
    atomicAdd(b + 3, v.w);
}

// ---------------------------------------------------------------------------
// Fused GCN GEMM using V_WMMA_F32_16X16X4_F32 (full fp32 precision):
//   Y = act((Xin * in_norm[:,None]) @ W + b) [* out_norm[:,None]]
// One wave computes a 16-row x 128-col strip: 8 v8f accumulators,
// K=128 consumed in 32 steps of K=4.
// A layout (16x4 f32): lane(0-15)=M, v0/v1 = K{0,1}; lanes 16-31 = K{2,3}.
// B layout (4x16 f32): lane(0-15)=N, v0/v1 = K{0,1}; lanes 16-31 = K{2,3}.
// C/D: VGPR j holds M=j (lanes 0-15) / M=j+8 (lanes 16-31), N=lane&15.
// ---------------------------------------------------------------------------
template <bool RELU, bool SCALE_OUT>
__global__ __launch_bounds__(128) void gcn_gemm_kernel(
    const float* __restrict__ Xin,      // [N,128] aggregated features
    const float* __restrict__ W,        // [128,128] row-major
    const float* __restrict__ bias,     // [128]
    const float* __restrict__ in_norm,  // [N]
    const float* __restrict__ out_norm, // [N]
    float* __restrict__ Y,              // [N,128]
    int num_tiles) {
    int wave = (int)(((long long)blockIdx.x * blockDim.x + threadIdx.x) >> 5);
    if (wave >= num_tiles) return;
    int lane = threadIdx.x & 31;
    int half = lane >> 4;   // 0: K pair {0,1}, 1: K pair {2,3}
    int lidx = lane & 15;   // M for A, N-within-tile for B/C/D
    int rowbase = wave * 16;
    int m = rowbase + lidx;
    float innrm = in_norm[m];

    v8f zero = {};
    v8f acc[8];
#pragma unroll
    for (int nt = 0; nt < 8; ++nt) acc[nt] = zero;

    const float* xrow = Xin + (size_t)m * 128;
    for (int kb = 0; kb < 32; ++kb) {
        int k0 = kb * 4 + half * 2;
        v2f a;
        a.x = xrow[k0] * innrm;
        a.y = xrow[k0 + 1] * innrm;
        const float* wrow = W + (size_t)k0 * 128;
#pragma unroll
        for (int nt = 0; nt < 8; ++nt) {
            int n = nt * 16 + lidx;
            v2f b;
            b.x = wrow[n];
            b.y = wrow[128 + n];
            acc[nt] = __builtin_amdgcn_wmma_f32_16x16x4_f32(
                false, a, false, b, (short)0, acc[nt], false, false);
        }
    }

    // epilogue: bias (+ReLU) (+out_norm), store
    float bn[8];
#pragma unroll
    for (int nt = 0; nt < 8; ++nt) bn[nt] = bias[nt * 16 + lidx];

#pragma unroll
    for (int j = 0; j < 8; ++j) {
        int mm = rowbase + j + 8 * half;
        float on = SCALE_OUT ? out_norm[mm] : 1.0f;
        float* yrow = Y + (size_t)mm * 128 + lidx;
#pragma unroll
        for (int nt = 0; nt < 8; ++nt) {
            float v = acc[nt][j] + bn[nt];
            if (RELU) v = fmaxf(v, 0.0f);
            if (SCALE_OUT) v = v * on;
            yrow[nt * 16] = v;
        }
    }
}

// ---------------------------------------------------------------------------
// Per-graph pooling: sums[g][f] += x[n][f]; counts[g] += 1
// ---------------------------------------------------------------------------
__global__ void pool_kernel(const float* __restrict__ x, const int* __restrict__ gid,
                            float* __restrict__ sums, float* __restrict__ counts,
                            long long NF) {
    long long i = (long long)blockIdx.x * blockDim.x + threadIdx.x;
    if (i >= NF) return;
    int n = (int)(i >> 7);
    int f = (int)(i & 127);
    int g = gid[n];
    atomicAdd(&sums[(size_t)g * 128 + f], x[i]);
    if (f == 0) atomicAdd(&counts[g], 1.0f);
}

// ---------------------------------------------------------------------------
// Final classifier: out[g][c] = [mean(sums[g]), perm[g]] @ Wc + bc
// G=128, P=64, C=10, K=192 — tiny.
// ---------------------------------------------------------------------------
__global__ void classify_kernel(const float* __restrict__ sums, const float* __restrict__ counts,
                                const float* __restrict__ perm, const float* __restrict__ Wc,
                                const float* __restrict__ bc, float* __restrict__ out,
                                int G, int P, int C) {
    int t = blockIdx.x * blockDim.x + threadIdx.x;
    if (t >= G * C) return;
    int g = t / C;
    int c = t % C;
    float inv = 1.0f / fmaxf(counts[g], 1.0f);
    float acc = bc[c];
    for (int k = 0; k < 128; ++k)
        acc += sums[(size_t)g * 128 + k] * inv * Wc[(size_t)k * C + c];
    for (int k = 0; k < P; ++k)
        acc += perm[(size_t)g * P + k] * Wc[(size_t)(128 + k) * C + c];
    out[(size_t)g * C + c] = acc;
}

// ---------------------------------------------------------------------------
extern "C" void kernel_launch(void* const* d_in, const int* in_sizes, int n_in,
                              void* d_out, int out_size, void* d_ws, size_t ws_size,
                              hipStream_t stream) {
    // inputs: h, perm_features, W1, b1, W2, b2, Wc, bc, src, dst, graph_ids
    const float* h    = (const float*)d_in[0];
    const float* perm = (const float*)d_in[1];
    const float* W1   = (const float*)d_in[2];
    const float* b1   = (const float*)d_in[3];
    const float* W2   = (const float*)d_in[4];
    const float* b2   = (const float*)d_in[5];
    const float* Wc   = (const float*)d_in[6];
    const float* bc   = (const float*)d_in[7];
    const int*   src  = (const int*)d_in[8];
    const int*   dst  = (const int*)d_in[9];
    const int*   gid  = (const int*)d_in[10];
    float* out = (float*)d_out;

    const int D = 128;                       // feature width (fixed)
    const int N = in_sizes[0] / D;           // 100000
    const int E = in_sizes[8];               // 1600000
    const int G = 128;                       // NUM_GRAPHS
    const int P = in_sizes[1] / G;           // 64
    const int C = out_size / G;              // 10
    const long long NF = (long long)N * D;

    // workspace layout (256B aligned slabs)
    size_t off = 0;
    auto alloc = [&](size_t bytes) -> size_t {
        size_t o = off;
        off = (off + bytes + 255) & ~(size_t)255;
        return o;
    };
    char* ws = (char*)d_ws;
    float* onorm  = (float*)(ws + alloc((size_t)N * 4));
    float* inorm  = (float*)(ws + alloc((size_t)N * 4));
    float* bufA   = (float*)(ws + alloc((size_t)NF * 4));   // scaled feats / gemm out
    float* bufB   = (float*)(ws + alloc((size_t)NF * 4));   // aggregation target
    float* sums   = (float*)(ws + alloc((size_t)G * D * 4));
    float* counts = (float*)(ws + alloc((size_t)G * 4));
    (void)ws_size;

    const int TPB = 256;
    const int num_tiles = N / 16;                       // N divisible by 16 (100000)
    const int gemm_blocks = (num_tiles + 3) / 4;        // 4 waves / 128-thread block
    const int agg_blocks = (E + 7) / 8;                 // 1 wave per edge, 8 waves / block

    // --- degrees & norms ---
    fill_zero_kernel<<<(int)((2LL * N + TPB - 1) / TPB), TPB, 0, stream>>>(onorm, 2LL * N);
    degree_kernel<<<(E + TPB - 1) / TPB, TPB, 0, stream>>>(src, dst, onorm, inorm, E);
    norm_kernel<<<(N + TPB - 1) / TPB, TPB, 0, stream>>>(onorm, inorm, N);

    // --- layer 1 ---
    scale_kernel<<<(int)((NF + TPB - 1) / TPB), TPB, 0, stream>>>(h, onorm, bufA, NF);
    fill_zero_kernel<<<(int)((NF + TPB - 1) / TPB), TPB, 0, stream>>>(bufB, NF);
    aggregate_kernel<<<agg_blocks, TPB, 0, stream>>>(bufA, src, dst, bufB, E);
    // bufA = relu((bufB*in_norm)@W1 + b1) * out_norm  (fused pre-scale for layer 2)
    gcn_gemm_kernel<true, true><<<gemm_blocks, 128, 0, stream>>>(
        bufB, W1, b1, inorm, onorm, bufA, num_tiles);

    // --- layer 2 ---
    fill_zero_kernel<<<(int)((NF + TPB - 1) / TPB), TPB, 0, stream>>>(bufB, NF);
    aggregate_kernel<<<agg_blocks, TPB, 0, stream>>>(bufA, src, dst, bufB, E);
    // bufA = (bufB*in_norm)@W2 + b2
    gcn_gemm_kernel<false, false><<<gemm_blocks, 128, 0, stream>>>(
        bufB, W2, b2, inorm, onorm, bufA, num_tiles);

    // --- pooling + classifier ---
    fill_zero_kernel<<<(G * D + G + TPB - 1) / TPB, TPB, 0, stream>>>(sums, (long long)G * D + G);
    pool_kernel<<<(int)((NF + TPB - 1) / TPB), TPB, 0, stream>>>(bufA, gid, sums, counts, NF);
    classify_kernel<<<(G * C + TPB - 1) / TPB, TPB, 0, stream>>>(
        sums, counts, perm, Wc, bc, out, G, P, C);
}